// GNNLongTermMemory_51032801411428
// MI455X (gfx1250) — compile-verified
//
#include <hip/hip_runtime.h>
#include <hip/hip_bf16.h>

typedef __attribute__((ext_vector_type(2))) float v2f;
typedef __attribute__((ext_vector_type(8))) float v8f;

static __device__ __forceinline__ v8f wmma_f32_4(v2f a, v2f b, v8f c) {
    return __builtin_amdgcn_wmma_f32_16x16x4_f32(
        false, a, false, b, (short)0, c, false, false);
}

// ---------------------------------------------------------------------------
// Kernel A: fused node projections via f32 WMMA, 2-deep software pipeline.
//   h_acc[n, :] = node_emb[n, :] @ Ws.T + bs
//   m_out[n, :] = node_emb[n, :] @ Wn.T + bn
// One wave32 per 16-node tile; 8 accumulator tiles (4 hid-tiles x {Ws,Wn});
// K=256 in 64 steps of 4 with V_WMMA_F32_16X16X4_F32. Fragments are
// prefetched TWO k-steps ahead in distinct buffers so both A and B loads
// get a full 8-WMMA group of slack -> counted s_wait_loadcnt, no drains.
// ---------------------------------------------------------------------------
__global__ __launch_bounds__(256) void gnn_proj_wmma(
    const float* __restrict__ node_emb,  // [N, 256]
    const float* __restrict__ Ws,        // [64, 256]
    const float* __restrict__ bs,        // [64]
    const float* __restrict__ Wn,        // [64, 256]
    const float* __restrict__ bn,        // [64]
    float* __restrict__ h_acc,           // [N, 64]
    float* __restrict__ m_out,           // [N, 64]
    int N)
{
    const int lane = threadIdx.x & 31;
    const int wave = threadIdx.x >> 5;
    const int tile = blockIdx.x * 8 + wave;     // 16-node tile
    const int ntiles = (N + 15) >> 4;
    if (tile >= ntiles) return;                 // wave-uniform exit

    const int row0 = tile * 16;
    const int half = lane >> 4;                 // 0: lanes 0-15, 1: lanes 16-31
    const int lmod = lane & 15;

    // A-fragment source row (clamped so EXEC stays all-ones through WMMA)
    int arow = row0 + lmod;
    if (arow >= N) arow = N - 1;
    const float* aptr = node_emb + (size_t)arow * 256 + half * 2;

    // Weight-row pointers for the 4 hidden 16-wide tiles (B fragments).
    const float* wsrow[4];
    const float* wnrow[4];
#pragma unroll
    for (int t = 0; t < 4; ++t) {
        wsrow[t] = Ws + (size_t)(t * 16 + lmod) * 256 + half * 2;
        wnrow[t] = Wn + (size_t)(t * 16 + lmod) * 256 + half * 2;
    }

    v8f acc[8];
#pragma unroll
    for (int i = 0; i < 8; ++i) acc[i] = (v8f)(0.0f);

    // ---- prologue: load k-steps 0 and 1 into stage buffers ----
    v2f a0, a1, b0[8], b1[8];
    a0 = *(const v2f*)(aptr);
    a1 = *(const v2f*)(aptr + 4);
#pragma unroll
    for (int t = 0; t < 4; ++t) {
        b0[2 * t]     = *(const v2f*)(wsrow[t]);
        b0[2 * t + 1] = *(const v2f*)(wnrow[t]);
        b1[2 * t]     = *(const v2f*)(wsrow[t] + 4);
        b1[2 * t + 1] = *(const v2f*)(wnrow[t] + 4);
    }

    // ---- main loop: 32 iterations, each processing 2 k-steps ----
    for (int kk = 0; kk < 64; kk += 2) {
        // prefetch k-step kk+2 (wrap keeps addresses in-bounds & branch-free)
        const int kb2 = ((kk + 2) & 63) * 4;
        v2f a2 = *(const v2f*)(aptr + kb2);
        v2f b2[8];
#pragma unroll
        for (int t = 0; t < 4; ++t) {
            b2[2 * t]     = *(const v2f*)(wsrow[t] + kb2);
            b2[2 * t + 1] = *(const v2f*)(wnrow[t] + kb2);
        }
        // compute k-step kk (fragments loaded 2 steps ago: full slack)
#pragma unroll
        for (int t = 0; t < 4; ++t) {
            acc[t]     = wmma_f32_4(a0, b0[2 * t],     acc[t]);
            acc[4 + t] = wmma_f32_4(a0, b0[2 * t + 1], acc[4 + t]);
        }

        // prefetch k-step kk+3
        const int kb3 = ((kk + 3) & 63) * 4;
        v2f a3 = *(const v2f*)(aptr + kb3);
        v2f b3[8];
#pragma unroll
        for (int t = 0; t < 4; ++t) {
            b3[2 * t]     = *(const v2f*)(wsrow[t] + kb3);
            b3[2 * t + 1] = *(const v2f*)(wnrow[t] + kb3);
        }
        // compute k-step kk+1
#pragma unroll
        for (int t = 0; t < 4; ++t) {
            acc[t]     = wmma_f32_4(a1, b1[2 * t],     acc[t]);
            acc[4 + t] = wmma_f32_4(a1, b1[2 * t + 1], acc[4 + t]);
        }

        // rotate stages
        a0 = a2;
        a1 = a3;
#pragma unroll
        for (int i = 0; i < 8; ++i) { b0[i] = b2[i]; b1[i] = b3[i]; }
    }

    // C/D layout: VGPR j holds row M=j (lanes 0-15) / M=j+8 (lanes 16-31),
    // column N = lane%16. Add per-column bias and store.
    const int mbase = row0 + half * 8;
#pragma unroll
    for (int t = 0; t < 4; ++t) {
        const int col = t * 16 + lmod;
        const float bsv = bs[col];
        const float bnv = bn[col];
#pragma unroll
        for (int j = 0; j < 8; ++j) {
            const int r = mbase + j;
            if (r < N) {
                h_acc[(size_t)r * 64 + col] = acc[t][j] + bsv;
                m_out[(size_t)r * 64 + col] = acc[4 + t][j] + bnv;
            }
        }
    }
}

// ---------------------------------------------------------------------------
// Kernel B: edge messages. 64 consecutive threads per edge (2 waves), one
// feature each: h_acc[dst] += edge_w[e] * m[src].  m and h_acc (25.6 MB
// each) are resident in the 192 MB L2, so gathers and the hardware f32
// atomics resolve at L2.
// ---------------------------------------------------------------------------
__global__ __launch_bounds__(256) void gnn_edge_scatter(
    const float* __restrict__ m,       // [N, 64]
    const float* __restrict__ edge_w,  // [E]
    const int* __restrict__ src,       // [E]
    const int* __restrict__ dst,       // [E]
    float* __restrict__ h_acc,         // [N, 64]
    int E)
{
    const long long t = (long long)blockIdx.x * blockDim.x + threadIdx.x;
    const long long e = t >> 6;
    const int f = (int)(t & 63);
    if (e >= E) return;
    const int s = src[e];
    const int d = dst[e];
    const float w = edge_w[e];
    const float val = w * m[(size_t)s * 64 + f];
    unsafeAtomicAdd(&h_acc[(size_t)d * 64 + f], val);  // global_atomic_add_f32
}

// ---------------------------------------------------------------------------
// Kernel C: readout. One wave32 per query; lane handles 2 features;
// relu then dot with Wr, shuffle-reduce, add br.
// ---------------------------------------------------------------------------
__global__ __launch_bounds__(256) void gnn_readout(
    const float* __restrict__ h_acc,  // [N, 64]
    const int* __restrict__ query,    // [Q]
    const float* __restrict__ Wr,     // [64]
    const float* __restrict__ br,     // [1]
    float* __restrict__ out,          // [Q]
    int Q)
{
    const int wid  = (int)((blockIdx.x * blockDim.x + threadIdx.x) >> 5);
    const int lane = threadIdx.x & 31;
    if (wid >= Q) return;  // wave-uniform
    const int node = query[wid];
    const float* h = h_acc + (size_t)node * 64 + lane * 2;
    const float* w = Wr + lane * 2;
    float a = fmaxf(h[0], 0.0f) * w[0] + fmaxf(h[1], 0.0f) * w[1];
#pragma unroll
    for (int off = 16; off > 0; off >>= 1)
        a += __shfl_xor(a, off, 32);
    if (lane == 0) out[wid] = a + br[0];
}

// ---------------------------------------------------------------------------
// Launch: inputs in setup_inputs() order.
//  0 node_emb [N*256] f32   1 edge_w [E] f32   2 Ws [64*256]   3 bs [64]
//  4 Wn [64*256]            5 bn [64]          6 Wr [64]       7 br [1]
//  8 src [E] i32            9 dst [E] i32     10 query [Q] i32
// Workspace: h_acc [N*64] f32, m [N*64] f32.
// ---------------------------------------------------------------------------
extern "C" void kernel_launch(void* const* d_in, const int* in_sizes, int n_in,
                              void* d_out, int out_size, void* d_ws, size_t ws_size,
                              hipStream_t stream) {
    const float* node_emb = (const float*)d_in[0];
    const float* edge_w   = (const float*)d_in[1];
    const float* Ws       = (const float*)d_in[2];
    const float* bs       = (const float*)d_in[3];
    const float* Wn       = (const float*)d_in[4];
    const float* bn       = (const float*)d_in[5];
    const float* Wr       = (const float*)d_in[6];
    const float* br       = (const float*)d_in[7];
    const int*   src      = (const int*)d_in[8];
    const int*   dst      = (const int*)d_in[9];
    const int*   query    = (const int*)d_in[10];

    const int N = in_sizes[0] / 256;
    const int E = in_sizes[1];
    const int Q = in_sizes[10];

    float* h_acc = (float*)d_ws;
    float* m     = h_acc + (size_t)N * 64;

    // Phase 1: fused projections (WMMA). 8 tiles (waves) per 256-thread block.
    const int ntiles  = (N + 15) / 16;
    const int blocksA = (ntiles + 7) / 8;
    gnn_proj_wmma<<<blocksA, 256, 0, stream>>>(node_emb, Ws, bs, Wn, bn,
                                               h_acc, m, N);

    // Phase 2: edge scatter-add (64 threads per edge).
    const long long workB = (long long)E * 64;
    const int blocksB = (int)((workB + 255) / 256);
    gnn_edge_scatter<<<blocksB, 256, 0, stream>>>(m, edge_w, src, dst, h_acc, E);

    // Phase 3: readout (1 wave per query).
    const int blocksC = (Q * 32 + 255) / 256;
    gnn_readout<<<blocksC, 256, 0, stream>>>(h_acc, query, Wr, br,
                                             (float*)d_out, Q);
}